// BaseModel_8504035246519
// MI455X (gfx1250) — compile-verified
//
#include <hip/hip_runtime.h>

// ---------------------------------------------------------------------------
// GCN (2x GCNConv + MLP head) for MI455X / gfx1250, wave32.
// Dense GEMMs use V_WMMA_F32_16X16X4_F32 (fp32 matrix path); the sparse
// aggregation (the real bottleneck: ~1 GB of mostly-L2 traffic, 2x102M f32
// atomics) uses one-lane-per-(edge,feature) global_atomic_add_f32.
// ---------------------------------------------------------------------------

typedef float v2f __attribute__((ext_vector_type(2)));
typedef float v8f __attribute__((ext_vector_type(8)));

#define FDIM 32  // hidden width of every layer after the first GEMM

// ---------- degree / normalization -----------------------------------------

__global__ void k_deg_init(float* __restrict__ dnorm, int n) {
    int i = blockIdx.x * blockDim.x + threadIdx.x;
    if (i < n) dnorm[i] = 1.0f;  // self-loop contributes 1 to every degree
}

__global__ void k_deg_edges(const int* __restrict__ col, float* __restrict__ dnorm, int ne) {
    int e = blockIdx.x * blockDim.x + threadIdx.x;
    if (e < ne) atomicAdd(&dnorm[col[e]], 1.0f);
}

__global__ void k_deg_rsqrt(float* __restrict__ dnorm, int n) {
    int i = blockIdx.x * blockDim.x + threadIdx.x;
    if (i < n) dnorm[i] = rsqrtf(dnorm[i]);  // deg>0 always (self loops)
}

// ---------- dense GEMM via f32 WMMA ----------------------------------------
// C[nrows x 32] = A[nrows x K] @ W[K x 32].  One wave per 16-row block,
// two 16x16 accumulators (cols 0-15 and 16-31), K stepped by 4.
// A-operand layout (ISA 7.12.2, 32-bit A 16x4): lanes 0-15 hold K=k0,k0+1;
// lanes 16-31 hold K=k0+2,k0+3; row = lane%16.

template <int K>
__global__ void k_gemm_wmma(const float* __restrict__ A, const float* __restrict__ W,
                            float* __restrict__ C, int nrows) {
    int wave = (int)((blockIdx.x * blockDim.x + threadIdx.x) >> 5);
    int lane = threadIdx.x & 31;
    int row0 = wave * 16;
    if (row0 >= nrows) return;          // wave-uniform: EXEC all-1 for WMMA
    const int half = lane >> 4;         // 0: lanes 0-15, 1: lanes 16-31
    const int l    = lane & 15;

    v8f c0 = {};
    v8f c1 = {};
    const float* __restrict__ arow = A + (size_t)(row0 + l) * K;

#pragma unroll
    for (int k0 = 0; k0 < K; k0 += 4) {
        const int ka = k0 + 2 * half;
        v2f a;
        a.x = arow[ka + 0];
        a.y = arow[ka + 1];
        v2f b0, b1;
        b0.x = W[(size_t)(ka + 0) * 32 + l];
        b0.y = W[(size_t)(ka + 1) * 32 + l];
        b1.x = W[(size_t)(ka + 0) * 32 + 16 + l];
        b1.y = W[(size_t)(ka + 1) * 32 + 16 + l];
        c0 = __builtin_amdgcn_wmma_f32_16x16x4_f32(false, a, false, b0, (short)0, c0, false, false);
        c1 = __builtin_amdgcn_wmma_f32_16x16x4_f32(false, a, false, b1, (short)0, c1, false, false);
    }

    // D layout: VGPR r -> row r (lanes 0-15) / row r+8 (lanes 16-31), col = l.
    float* __restrict__ crow = C + (size_t)(row0 + 8 * half) * 32 + l;
#pragma unroll
    for (int r = 0; r < 8; ++r) {
        crow[(size_t)r * 32]      = c0[r];
        crow[(size_t)r * 32 + 16] = c1[r];
    }
}

// ---------- sparse aggregation ---------------------------------------------

// Seed accumulator with the self-loop term: agg[i] = dinv[i]^2 * h[i]
__global__ void k_self_seed(const float* __restrict__ h, const float* __restrict__ dinv,
                            float* __restrict__ agg, int n) {
    long long t = (long long)blockIdx.x * blockDim.x + threadIdx.x;
    int i = (int)(t >> 5);
    int f = (int)(t & 31);
    if (i < n) {
        float di = dinv[i];
        agg[(size_t)i * 32 + f] = di * di * h[(size_t)i * 32 + f];
    }
}

// One lane per (edge, feature): coalesced gather of h[src], L2-resident atomics.
__global__ void k_scatter_edges(const int* __restrict__ row, const int* __restrict__ col,
                                const float* __restrict__ dinv, const float* __restrict__ h,
                                float* __restrict__ agg, int ne) {
    long long t = (long long)blockIdx.x * blockDim.x + threadIdx.x;
    int e = (int)(t >> 5);
    int f = (int)(t & 31);
    if (e >= ne) return;
    int s = row[e];
    int d = col[e];
    float w = dinv[s] * dinv[d];
    atomicAdd(&agg[(size_t)d * 32 + f], w * h[(size_t)s * 32 + f]);
}

__global__ void k_bias_relu(const float* __restrict__ agg, const float* __restrict__ b,
                            float* __restrict__ h, long long total) {
    long long t = (long long)blockIdx.x * blockDim.x + threadIdx.x;
    if (t < total) {
        int f = (int)(t & 31);
        h[t] = fmaxf(agg[t] + b[f], 0.0f);
    }
}

// ---------- fused MLP head: out = relu(H @ Wl1 + bl1) @ Wl4 + bl4 ----------

__global__ void k_head_wmma(const float* __restrict__ H, const float* __restrict__ Wl1,
                            const float* __restrict__ bl1, const float* __restrict__ Wl4,
                            const float* __restrict__ bl4, float* __restrict__ out, int nrows) {
    int wave = (int)((blockIdx.x * blockDim.x + threadIdx.x) >> 5);
    int lane = threadIdx.x & 31;
    int row0 = wave * 16;
    if (row0 >= nrows) return;
    const int half = lane >> 4;
    const int l    = lane & 15;

    v8f c0 = {};
    v8f c1 = {};
    const float* __restrict__ arow = H + (size_t)(row0 + l) * 32;

#pragma unroll
    for (int k0 = 0; k0 < 32; k0 += 4) {
        const int ka = k0 + 2 * half;
        v2f a;
        a.x = arow[ka + 0];
        a.y = arow[ka + 1];
        v2f b0, b1;
        b0.x = Wl1[(size_t)(ka + 0) * 32 + l];
        b0.y = Wl1[(size_t)(ka + 1) * 32 + l];
        b1.x = Wl1[(size_t)(ka + 0) * 32 + 16 + l];
        b1.y = Wl1[(size_t)(ka + 1) * 32 + 16 + l];
        c0 = __builtin_amdgcn_wmma_f32_16x16x4_f32(false, a, false, b0, (short)0, c0, false, false);
        c1 = __builtin_amdgcn_wmma_f32_16x16x4_f32(false, a, false, b1, (short)0, c1, false, false);
    }

    const float w0 = Wl4[l];
    const float w1 = Wl4[16 + l];
    const float bias0 = bl1[l];
    const float bias1 = bl1[16 + l];
    const float b4 = bl4[0];

#pragma unroll
    for (int r = 0; r < 8; ++r) {
        float t0 = fmaxf(c0[r] + bias0, 0.0f);
        float t1 = fmaxf(c1[r] + bias1, 0.0f);
        float p = t0 * w0 + t1 * w1;
        // reduce the 16 columns held by each lane-half (xor masks stay in-half)
        p += __shfl_xor(p, 1, 32);
        p += __shfl_xor(p, 2, 32);
        p += __shfl_xor(p, 4, 32);
        p += __shfl_xor(p, 8, 32);
        if (l == 0) out[row0 + r + 8 * half] = p + b4;
    }
}

// ---------------------------------------------------------------------------

extern "C" void kernel_launch(void* const* d_in, const int* in_sizes, int n_in,
                              void* d_out, int out_size, void* d_ws, size_t ws_size,
                              hipStream_t stream) {
    (void)n_in; (void)out_size; (void)ws_size;

    const float* x   = (const float*)d_in[0];   // [N,16]
    const int*   ei  = (const int*)  d_in[1];   // [2,E]
    const float* W1  = (const float*)d_in[2];   // [16,32]
    const float* b1  = (const float*)d_in[3];   // [32]
    const float* W2  = (const float*)d_in[4];   // [32,32]
    const float* b2  = (const float*)d_in[5];   // [32]
    const float* Wl1 = (const float*)d_in[6];   // [32,32]
    const float* bl1 = (const float*)d_in[7];   // [32]
    const float* Wl4 = (const float*)d_in[8];   // [32,1]
    const float* bl4 = (const float*)d_in[9];   // [1]
    float* out = (float*)d_out;                 // [N,1]

    const int N = in_sizes[0] / 16;
    const int E = in_sizes[1] / 2;
    const int* e_row = ei;       // sources j
    const int* e_col = ei + E;   // targets i

    // workspace layout (floats): dnorm[N] | A[N*32] | B[N*32]
    float* dnorm = (float*)d_ws;
    float* A = dnorm + (((size_t)N + 511) / 512) * 512;
    float* B = A + (size_t)N * 32;

    const int TB = 256;
    const unsigned gN    = (unsigned)((N + TB - 1) / TB);
    const unsigned gE    = (unsigned)((E + TB - 1) / TB);
    const unsigned gNF   = (unsigned)(((long long)N * 32 + TB - 1) / TB);
    const unsigned gEF   = (unsigned)(((long long)E * 32 + TB - 1) / TB);
    const int waves      = N / 16;                       // N % 16 == 0 here
    const unsigned gWmma = (unsigned)((waves * 32 + TB - 1) / TB);
    const long long NF   = (long long)N * 32;

    // --- normalization: dinv = rsqrt(1 + indegree) ---
    k_deg_init<<<gN, TB, 0, stream>>>(dnorm, N);
    k_deg_edges<<<gE, TB, 0, stream>>>(e_col, dnorm, E);
    k_deg_rsqrt<<<gN, TB, 0, stream>>>(dnorm, N);

    // --- layer 1: h = relu(Ahat @ (x@W1) + b1) ---
    k_gemm_wmma<16><<<gWmma, TB, 0, stream>>>(x, W1, A, N);
    k_self_seed<<<gNF, TB, 0, stream>>>(A, dnorm, B, N);
    k_scatter_edges<<<gEF, TB, 0, stream>>>(e_row, e_col, dnorm, A, B, E);
    k_bias_relu<<<gNF, TB, 0, stream>>>(B, b1, B, NF);

    // --- layer 2: h = relu(Ahat @ (h@W2) + b2) ---
    k_gemm_wmma<32><<<gWmma, TB, 0, stream>>>(B, W2, A, N);
    k_self_seed<<<gNF, TB, 0, stream>>>(A, dnorm, B, N);
    k_scatter_edges<<<gEF, TB, 0, stream>>>(e_row, e_col, dnorm, A, B, E);
    k_bias_relu<<<gNF, TB, 0, stream>>>(B, b2, B, NF);

    // --- head: out = relu(h@Wl1 + bl1) @ Wl4 + bl4 ---
    k_head_wmma<<<gWmma, TB, 0, stream>>>(B, Wl1, bl1, Wl4, bl4, out, N);
}